// Net_9912784520004
// MI455X (gfx1250) — compile-verified
//
#include <hip/hip_runtime.h>

// ---------------------------------------------------------------------------
// Submanifold sparse conv stack for MI455X (gfx1250, wave32).
// Rulebook build + per-layer (transpose-W, gather-WMMA-GEMM).
// Matrix core: V_WMMA_F32_16X16X4_F32 (fp32 in/out, matches reference
// precision). Working set (~200MB ws, 80MB feature buffers) is L2-resident,
// so the GEMM is compute/issue bound: each wave owns a 32x64 (or 32x32)
// output tile -> 8 (4) independent WMMA accumulator chains, 0.75 (1.0)
// vector loads per WMMA, gathered A fragments reused across 4 column tiles.
// ---------------------------------------------------------------------------

typedef __attribute__((ext_vector_type(2))) float v2f;
typedef __attribute__((ext_vector_type(8))) float v8f;

#define DD   41
#define HH   400
#define WWW  352
#define DHW  (DD * HH * WWW)          // 5,772,800
#define NPTS 80000
#define NROWT (NPTS / 32)             // 2500 row tiles of 32, exact

// ---------------------------- rulebook kernels -----------------------------

__global__ void fill_grid_kernel(int* __restrict__ grid) {
    int i = blockIdx.x * blockDim.x + threadIdx.x;
    if (i < DHW) grid[i] = -1;
}

__global__ void scatter_grid_kernel(const int* __restrict__ coors,
                                    int* __restrict__ grid) {
    int i = blockIdx.x * blockDim.x + threadIdx.x;
    if (i < NPTS) {
        int z = coors[4 * i + 1];
        int y = coors[4 * i + 2];
        int x = coors[4 * i + 3];
        grid[z * (HH * WWW) + y * WWW + x] = i;
    }
}

__global__ void build_idx_kernel(const int* __restrict__ coors,
                                 const int* __restrict__ grid,
                                 int* __restrict__ idx) {
    int t = blockIdx.x * blockDim.x + threadIdx.x;
    if (t >= 27 * NPTS) return;
    int k = t / NPTS;
    int i = t - k * NPTS;
    int dz = k / 9 - 1;
    int dy = (k / 3) % 3 - 1;
    int dx = k % 3 - 1;
    int nz = coors[4 * i + 1] + dz;
    int ny = coors[4 * i + 2] + dy;
    int nx = coors[4 * i + 3] + dx;
    int v = -1;
    if (nz >= 0 && nz < DD && ny >= 0 && ny < HH && nx >= 0 && nx < WWW) {
        v = grid[nz * (HH * WWW) + ny * WWW + nx];   // may itself be -1
    }
    idx[t] = v;
}

// Pad [N,3] features to [N,4] (K of fp32 WMMA is 4).
__global__ void pad_features_kernel(const float* __restrict__ f,
                                    float* __restrict__ out) {
    int t = blockIdx.x * blockDim.x + threadIdx.x;
    if (t >= NPTS * 4) return;
    int i = t >> 2, c = t & 3;
    out[t] = (c < 3) ? f[i * 3 + c] : 0.0f;
}

// W[k][ci][co] -> Wt[k][co][ciP]  (zero-pad ci >= cin)
__global__ void transpose_w_kernel(const float* __restrict__ w,
                                   float* __restrict__ wt,
                                   int cin, int cinP, int cout) {
    int t = blockIdx.x * blockDim.x + threadIdx.x;
    int per = cinP * cout;
    if (t >= 27 * per) return;
    int k   = t / per;
    int rem = t - k * per;
    int co  = rem / cinP;
    int ci  = rem - co * cinP;
    wt[t] = (ci < cin) ? w[(k * cin + ci) * cout + co] : 0.0f;
}

// ------------------------- gather-GEMM (WMMA f32) --------------------------
// A fragment (16x4 f32): lanes 0-15 rows M=lane%16, VGPR0/1 = K0/K1;
// lanes 16-31 same rows, K2/K3 -> one 8B load per lane per K-step.
// B fragment mirrors with lane = column N, from pre-transposed Wt[k][co][ci].
// Wave tile: 32 rows (2 row fragments) x COLW cols (COLW/16 col tiles).

template <int CINP, int COUT, int COLW>
__global__ __launch_bounds__(256) void subm_gemm_kernel(
    const float* __restrict__ feat, const float* __restrict__ wt,
    const int* __restrict__ idx, float* __restrict__ out) {
    constexpr int COLT = COUT / COLW;               // col tiles in grid
    constexpr int CT   = COLW / 16;                 // 16-col sub-tiles per wave
    const int wave = (blockIdx.x * 256 + (int)threadIdx.x) >> 5;
    const int lane = threadIdx.x & 31;
    const int rowTile = wave / COLT;
    const int colTile = wave - rowTile * COLT;
    if (rowTile >= NROWT) return;                   // wave-uniform exit

    const int rowBase = rowTile << 5;               // 32 rows per wave
    const int coBase  = colTile * COLW;
    const int lm = lane & 15;
    const int hi = lane >> 4;                        // K half: 0 -> K0,1 ; 1 -> K2,3
    const int koff = 2 * hi;

    v8f acc[2][CT];
#pragma unroll
    for (int f = 0; f < 2; ++f)
#pragma unroll
        for (int t = 0; t < CT; ++t)
            acc[f][t] = (v8f){0.f, 0.f, 0.f, 0.f, 0.f, 0.f, 0.f, 0.f};

#pragma unroll 1
    for (int k = 0; k < 27; ++k) {
        const int r0 = idx[k * NPTS + rowBase + lm];
        const int r1 = idx[k * NPTS + rowBase + 16 + lm];
        const int rs0 = (r0 < 0) ? 0 : r0;          // keep address in-bounds
        const int rs1 = (r1 < 0) ? 0 : r1;
        const float* aRow0 = feat + (size_t)rs0 * CINP + koff;
        const float* aRow1 = feat + (size_t)rs1 * CINP + koff;
        const float* bp =
            wt + ((size_t)(k * COUT) + coBase + lm) * CINP + koff;

#pragma unroll 4
        for (int ci = 0; ci < CINP; ci += 4) {
            v2f a0 = *(const v2f*)(aRow0 + ci);
            v2f a1 = *(const v2f*)(aRow1 + ci);
            if (r0 < 0) { a0.x = 0.0f; a0.y = 0.0f; }
            if (r1 < 0) { a1.x = 0.0f; a1.y = 0.0f; }
            v2f b[CT];
#pragma unroll
            for (int t = 0; t < CT; ++t)
                b[t] = *(const v2f*)(bp + (size_t)(16 * t) * CINP + ci);
#pragma unroll
            for (int t = 0; t < CT; ++t) {
                acc[0][t] = __builtin_amdgcn_wmma_f32_16x16x4_f32(
                    false, a0, false, b[t], (short)0, acc[0][t], false, false);
                acc[1][t] = __builtin_amdgcn_wmma_f32_16x16x4_f32(
                    false, a1, false, b[t], (short)0, acc[1][t], false, false);
            }
        }
    }

    // C/D layout: VGPR m -> row (m + 8*hi), lane%16 -> column.
#pragma unroll
    for (int f = 0; f < 2; ++f) {
#pragma unroll
        for (int m = 0; m < 8; ++m) {
            const int row = rowBase + 16 * f + m + 8 * hi;
#pragma unroll
            for (int t = 0; t < CT; ++t)
                out[(size_t)row * COUT + coBase + 16 * t + lm] = acc[f][t][m];
        }
    }
}

// ------------------------------- host side ---------------------------------

template <int CINP, int COUT>
static inline void launch_gemm(const float* feat, const float* wt,
                               const int* idx, float* out, hipStream_t s) {
    constexpr int COLW = (COUT % 64 == 0) ? 64 : 32;
    const int waves  = NROWT * (COUT / COLW);
    const int blocks = (waves + 7) / 8;              // 8 waves / 256-thread block
    subm_gemm_kernel<CINP, COUT, COLW><<<blocks, 256, 0, s>>>(feat, wt, idx, out);
}

extern "C" void kernel_launch(void* const* d_in, const int* in_sizes, int n_in,
                              void* d_out, int out_size, void* d_ws, size_t ws_size,
                              hipStream_t stream) {
    (void)in_sizes; (void)n_in; (void)out_size; (void)ws_size;

    const float* features = (const float*)d_in[0];   // [N,3]
    const int*   coors    = (const int*)d_in[1];     // [N,4]
    // d_in[2] = batch_size (unused)
    const float* Wl[14];
    for (int l = 0; l < 14; ++l) Wl[l] = (const float*)d_in[3 + l];

    // ---- workspace layout (all offsets 256B aligned) ----
    char* ws = (char*)d_ws;
    size_t off = 0;
    int*   grid = (int*)(ws + off);   off += (size_t)DHW * 4;            // 23,091,200
    int*   idx  = (int*)(ws + off);   off += (size_t)27 * NPTS * 4;      //  8,640,000
    float* bufA = (float*)(ws + off); off += (size_t)NPTS * 256 * 4;     // 81,920,000
    float* bufB = (float*)(ws + off); off += (size_t)NPTS * 256 * 4;     // 81,920,000
    float* wt   = (float*)(ws + off); off += (size_t)27 * 256 * 256 * 4; //  7,077,888
    // total ~202.7 MB

    // ---- rulebook ----
    fill_grid_kernel<<<(DHW + 255) / 256, 256, 0, stream>>>(grid);
    scatter_grid_kernel<<<(NPTS + 255) / 256, 256, 0, stream>>>(coors, grid);
    build_idx_kernel<<<(27 * NPTS + 255) / 256, 256, 0, stream>>>(coors, grid, idx);

    // ---- pad features [N,3] -> [N,4] ----
    pad_features_kernel<<<(NPTS * 4 + 255) / 256, 256, 0, stream>>>(features, bufA);

    static const int CIN[14]  = {3, 64, 64, 96, 96, 128, 128, 160, 160, 192, 192, 224, 224, 256};
    static const int CINP[14] = {4, 64, 64, 96, 96, 128, 128, 160, 160, 192, 192, 224, 224, 256};
    static const int COUT[14] = {64, 64, 96, 96, 128, 128, 160, 160, 192, 192, 224, 224, 256, 256};

    for (int l = 0; l < 14; ++l) {
        // transpose / pad weights for this layer
        const int total = 27 * CINP[l] * COUT[l];
        transpose_w_kernel<<<(total + 255) / 256, 256, 0, stream>>>(
            Wl[l], wt, CIN[l], CINP[l], COUT[l]);

        const float* fin = (l % 2 == 0) ? bufA : bufB;
        float* fout = (l == 13) ? (float*)d_out : ((l % 2 == 0) ? bufB : bufA);

        switch (l) {
            case 0:  launch_gemm<4,   64 >(fin, wt, idx, fout, stream); break;
            case 1:  launch_gemm<64,  64 >(fin, wt, idx, fout, stream); break;
            case 2:  launch_gemm<64,  96 >(fin, wt, idx, fout, stream); break;
            case 3:  launch_gemm<96,  96 >(fin, wt, idx, fout, stream); break;
            case 4:  launch_gemm<96,  128>(fin, wt, idx, fout, stream); break;
            case 5:  launch_gemm<128, 128>(fin, wt, idx, fout, stream); break;
            case 6:  launch_gemm<128, 160>(fin, wt, idx, fout, stream); break;
            case 7:  launch_gemm<160, 160>(fin, wt, idx, fout, stream); break;
            case 8:  launch_gemm<160, 192>(fin, wt, idx, fout, stream); break;
            case 9:  launch_gemm<192, 192>(fin, wt, idx, fout, stream); break;
            case 10: launch_gemm<192, 224>(fin, wt, idx, fout, stream); break;
            case 11: launch_gemm<224, 224>(fin, wt, idx, fout, stream); break;
            case 12: launch_gemm<224, 256>(fin, wt, idx, fout, stream); break;
            case 13: launch_gemm<256, 256>(fin, wt, idx, fout, stream); break;
        }
    }
}